// SSM_28578712387730
// MI455X (gfx1250) — compile-verified
//
#include <hip/hip_runtime.h>
#include <math.h>

typedef __attribute__((ext_vector_type(2))) float v2f;
typedef __attribute__((ext_vector_type(8))) float v8f;
typedef unsigned int u32x4 __attribute__((ext_vector_type(4)));
typedef int i32x4 __attribute__((ext_vector_type(4)));
typedef int i32x8 __attribute__((ext_vector_type(8)));

#define B_SZ   2
#define LSEQ   2048
#define DDIM   2048
#define DRANK  128
#define NST    16
#define PROJW  160            // DR + 2N
#define BL     (B_SZ * LSEQ)  // 4096 rows
#define NCHUNK 16
#define CHUNK  (LSEQ / NCHUNK) // 128

#if __has_builtin(__builtin_amdgcn_tensor_load_to_lds)
#define HAVE_TDM 1
#endif

// ---------------------------------------------------------------------------
// TDM-staged tile copy: proj[r0..r0+CHUNK) rows, WIDTH floats/row starting at
// column DRANK, row stride PROJW floats -> dense LDS tile [CHUNK][WIDTH].
// One wave issues the DMA; whole block barriers on completion.
// ---------------------------------------------------------------------------
template <int WIDTH>
__device__ inline void load_bc_tile(float* sdst, const float* gsrc) {
#ifdef HAVE_TDM
  if (threadIdx.x < 32) {
    const unsigned lds = (unsigned)(uintptr_t)sdst;            // LDS byte addr
    const unsigned long long ga = (unsigned long long)(uintptr_t)gsrc;
    // D# group0: count=1 | lds_addr | global_addr[56:0] | type=2
    u32x4 g0 = { 1u, lds, (unsigned)ga, (unsigned)(ga >> 32) | (2u << 30) };
    // D# group1: data_size=2 (4B); tensor_dim0=WIDTH; tensor_dim1=CHUNK;
    //            tile_dim0=WIDTH; tile_dim1=CHUNK; dim0_stride=PROJW
    i32x8 g1 = { 0x20000, WIDTH << 16, CHUNK << 16, WIDTH << 16,
                 CHUNK, PROJW, 0, 0 };
    i32x4 z4 = { 0, 0, 0, 0 };
#if __clang_major__ >= 23
    i32x8 z8 = { 0, 0, 0, 0, 0, 0, 0, 0 };
    __builtin_amdgcn_tensor_load_to_lds(g0, g1, z4, z4, z8, 0);
#else
    __builtin_amdgcn_tensor_load_to_lds(g0, g1, z4, z4, 0);
#endif
    __builtin_amdgcn_s_wait_tensorcnt(0);
  }
  __syncthreads();
#else
  for (int i = threadIdx.x; i < CHUNK * WIDTH; i += blockDim.x)
    sdst[i] = gsrc[(i / WIDTH) * PROJW + (i % WIDTH)];
  __syncthreads();
#endif
}

// ---------------------------------------------------------------------------
// Kernel 1: proj = x @ W_xproj^T + b_xproj   (4096x2048)x(2048x160)
// One wave per 16-row M tile; each wave produces all 10 N tiles (160 cols).
// fp32 WMMA 16x16x4: A lane layout m=lane&15, k-pair=(lane>>4)*2.
// ---------------------------------------------------------------------------
__global__ __launch_bounds__(256) void proj_gemm_k(
    const float* __restrict__ x, const float* __restrict__ W,
    const float* __restrict__ bias, float* __restrict__ proj) {
  const int lane  = threadIdx.x & 31;
  const int wave  = threadIdx.x >> 5;
  const int mtile = blockIdx.x * (blockDim.x >> 5) + wave;   // 0..255
  const int m0    = mtile * 16;
  const int mrow  = lane & 15;
  const int hi    = lane >> 4;
  const int koff  = hi * 2;

  v8f acc[10] = {};
  const float* xrow = x + (size_t)(m0 + mrow) * DDIM + koff;
  __builtin_prefetch(xrow, 0, 3);   // global_prefetch_b8 warm-up

  for (int k = 0; k < DDIM; k += 4) {
    v2f a = *(const v2f*)(xrow + k);
#pragma unroll
    for (int j = 0; j < 10; ++j) {
      v2f b = *(const v2f*)(W + (size_t)(j * 16 + mrow) * DDIM + k + koff);
      acc[j] = __builtin_amdgcn_wmma_f32_16x16x4_f32(
          false, a, false, b, (short)0, acc[j], false, false);
    }
  }
#pragma unroll
  for (int j = 0; j < 10; ++j) {
    const int col = j * 16 + (lane & 15);
    const float bv = bias[col];
#pragma unroll
    for (int i = 0; i < 8; ++i) {
      const int row = m0 + i + 8 * hi;
      proj[(size_t)row * PROJW + col] = acc[j][i] + bv;
    }
  }
}

// ---------------------------------------------------------------------------
// Kernel 2: dt = softplus(dt_r @ W_dt^T + b_dt)  (4096x128)x(128x2048)
// One wave per (16-row M tile, 128-col N group) => 4096 waves, K-loop = 32.
// ---------------------------------------------------------------------------
__global__ __launch_bounds__(256) void dt_gemm_k(
    const float* __restrict__ proj, const float* __restrict__ Wdt,
    const float* __restrict__ bdt, float* __restrict__ dt) {
  const int lane = threadIdx.x & 31;
  const int wave = threadIdx.x >> 5;
  const int wg   = blockIdx.x * (blockDim.x >> 5) + wave;    // 0..4095
  const int mtile = wg >> 4;          // 0..255
  const int ngrp  = wg & 15;          // 0..15 -> 128 cols each
  const int m0 = mtile * 16;
  const int n0 = ngrp * 128;
  const int mrow = lane & 15;
  const int hi   = lane >> 4;
  const int koff = hi * 2;

  v8f acc[8] = {};
  const float* arow = proj + (size_t)(m0 + mrow) * PROJW + koff;
  for (int k = 0; k < DRANK; k += 4) {
    v2f a = *(const v2f*)(arow + k);
#pragma unroll
    for (int j = 0; j < 8; ++j) {
      const int col = n0 + j * 16 + mrow;
      v2f b = *(const v2f*)(Wdt + (size_t)col * DRANK + k + koff);
      acc[j] = __builtin_amdgcn_wmma_f32_16x16x4_f32(
          false, a, false, b, (short)0, acc[j], false, false);
    }
  }
#pragma unroll
  for (int j = 0; j < 8; ++j) {
    const int col = n0 + j * 16 + (lane & 15);
    const float bv = bdt[col];
#pragma unroll
    for (int i = 0; i < 8; ++i) {
      const int row = m0 + i + 8 * hi;
      float z = acc[j][i] + bv;
      float sp = (z > 20.f) ? z : log1pf(__expf(z));   // softplus
      dt[(size_t)row * DDIM + col] = sp;
    }
  }
}

// ---------------------------------------------------------------------------
// Kernel 3a: per-(b,d,chunk) local scan from h=0; emit (P = prod dA, S = h_end)
// 65536 threads; block covers 256 consecutive d of one (b,chunk).
// B-slice for the chunk staged to LDS via TDM (128 rows x 16 floats).
// ---------------------------------------------------------------------------
__global__ __launch_bounds__(256) void scan_local_k(
    const float* __restrict__ x, const float* __restrict__ proj,
    const float* __restrict__ dt, const float* __restrict__ A_log,
    float* __restrict__ P, float* __restrict__ S) {
  __shared__ float sB[CHUNK * NST];                    // 8 KB
  const int bc    = blockIdx.x >> 3;                   // uniform: (b*16+chunk)
  const int b     = bc >> 4;
  const int chunk = bc & 15;
  const int idx   = blockIdx.x * blockDim.x + threadIdx.x;
  const int d     = idx & (DDIM - 1);
  const int l0    = chunk * CHUNK;
  const int r0    = b * LSEQ + l0;

  load_bc_tile<NST>(sB, proj + (size_t)r0 * PROJW + DRANK);

  float a[NST], h[NST], p[NST];
#pragma unroll
  for (int n = 0; n < NST; ++n) {
    a[n] = -__expf(A_log[d * NST + n]);
    h[n] = 0.f;
    p[n] = 1.f;
  }
  for (int li = 0; li < CHUNK; ++li) {
    const int r = r0 + li;
    const float dtv = dt[(size_t)r * DDIM + d];
    const float xv  = x[(size_t)r * DDIM + d];
    const float u   = dtv * xv;
    const float* Brow = sB + li * NST;
#pragma unroll
    for (int n = 0; n < NST; ++n) {
      const float dA = __expf(dtv * a[n]);
      h[n] = dA * h[n] + u * Brow[n];
      p[n] *= dA;
    }
  }
  float* Pp = P + (size_t)idx * NST;
  float* Sp = S + (size_t)idx * NST;
#pragma unroll
  for (int n = 0; n < NST; ++n) { Pp[n] = p[n]; Sp[n] = h[n]; }
}

// ---------------------------------------------------------------------------
// Kernel 3b: cross-chunk carry scan. One thread per (b,d,n) = 65536 threads.
// carry[chunk] = state entering chunk; recurrence carry' = P*carry + S.
// ---------------------------------------------------------------------------
__global__ __launch_bounds__(256) void carry_scan_k(
    const float* __restrict__ P, const float* __restrict__ S,
    float* __restrict__ Carry) {
  const int j = blockIdx.x * blockDim.x + threadIdx.x;  // 0..65535
  const int n = j & (NST - 1);
  const int d = (j >> 4) & (DDIM - 1);
  const int b = j >> 15;
  float c = 0.f;
  for (int chunk = 0; chunk < NCHUNK; ++chunk) {
    const size_t o = ((size_t)(((b * 16 + chunk) << 11) | d) << 4) | n;
    Carry[o] = c;
    c = P[o] * c + S[o];
  }
}

// ---------------------------------------------------------------------------
// Kernel 3c: replay local scan from carry, produce y = sum_n h*C.
// B+C slices (128 rows x 32 floats) staged to LDS via TDM.
// ---------------------------------------------------------------------------
__global__ __launch_bounds__(256) void scan_final_k(
    const float* __restrict__ x, const float* __restrict__ proj,
    const float* __restrict__ dt, const float* __restrict__ A_log,
    const float* __restrict__ Carry, float* __restrict__ y) {
  __shared__ float sBC[CHUNK * 2 * NST];               // 16 KB
  const int bc    = blockIdx.x >> 3;                   // uniform
  const int b     = bc >> 4;
  const int chunk = bc & 15;
  const int idx   = blockIdx.x * blockDim.x + threadIdx.x;
  const int d     = idx & (DDIM - 1);
  const int l0    = chunk * CHUNK;
  const int r0    = b * LSEQ + l0;

  load_bc_tile<2 * NST>(sBC, proj + (size_t)r0 * PROJW + DRANK);

  float a[NST], h[NST];
  const float* Cin = Carry + (size_t)idx * NST;
#pragma unroll
  for (int n = 0; n < NST; ++n) {
    a[n] = -__expf(A_log[d * NST + n]);
    h[n] = Cin[n];
  }
  for (int li = 0; li < CHUNK; ++li) {
    const int r = r0 + li;
    const float dtv = dt[(size_t)r * DDIM + d];
    const float xv  = x[(size_t)r * DDIM + d];
    const float u   = dtv * xv;
    const float* Brow = sBC + li * (2 * NST);
    const float* Crow = Brow + NST;
    float acc = 0.f;
#pragma unroll
    for (int n = 0; n < NST; ++n) {
      const float dA = __expf(dtv * a[n]);
      h[n] = dA * h[n] + u * Brow[n];
      acc += h[n] * Crow[n];
    }
    y[(size_t)r * DDIM + d] = acc;
  }
}

// ---------------------------------------------------------------------------
extern "C" void kernel_launch(void* const* d_in, const int* in_sizes, int n_in,
                              void* d_out, int out_size, void* d_ws, size_t ws_size,
                              hipStream_t stream) {
  const float* x       = (const float*)d_in[0];
  const float* W_xproj = (const float*)d_in[1];
  const float* b_xproj = (const float*)d_in[2];
  const float* W_dt    = (const float*)d_in[3];
  const float* b_dt    = (const float*)d_in[4];
  const float* A_log   = (const float*)d_in[5];
  float* y = (float*)d_out;

  char* ws = (char*)d_ws;
  float* proj  = (float*)ws;                                  // 4096*160
  ws += (size_t)BL * PROJW * sizeof(float);
  float* dt    = (float*)ws;                                  // 4096*2048
  ws += (size_t)BL * DDIM * sizeof(float);
  float* P     = (float*)ws;                                  // 32*2048*16
  ws += (size_t)B_SZ * NCHUNK * DDIM * NST * sizeof(float);
  float* S     = (float*)ws;
  ws += (size_t)B_SZ * NCHUNK * DDIM * NST * sizeof(float);
  float* Carry = (float*)ws;

  // GEMM1: 256 M-tiles, 8 waves/block -> 32 blocks
  proj_gemm_k<<<dim3(32), dim3(256), 0, stream>>>(x, W_xproj, b_xproj, proj);
  // GEMM2: 4096 waves, 8 waves/block -> 512 blocks
  dt_gemm_k<<<dim3(512), dim3(256), 0, stream>>>(proj, W_dt, b_dt, dt);
  // Chunked scan: 65536 threads each pass
  scan_local_k<<<dim3(256), dim3(256), 0, stream>>>(x, proj, dt, A_log, P, S);
  carry_scan_k<<<dim3(256), dim3(256), 0, stream>>>(P, S, Carry);
  scan_final_k<<<dim3(256), dim3(256), 0, stream>>>(x, proj, dt, A_log, Carry, y);
}